// NNUEModel_63814624084084
// MI455X (gfx1250) — compile-verified
//
#include <hip/hip_runtime.h>

#define B_SZ   2048
#define K_FT   32
#define D_FT   2056   // L1 + P
#define L1     2048
#define NLS    8

typedef float v2f __attribute__((ext_vector_type(2)));
typedef float v8f __attribute__((ext_vector_type(8)));

// ---------------------------------------------------------------------------
// Kernel 0: zero the per-stack bucket counters (graph-replay safe)
// ---------------------------------------------------------------------------
__global__ void nnue_init_counts(int* counts) {
    if (threadIdx.x < NLS) counts[threadIdx.x] = 0;
}

// ---------------------------------------------------------------------------
// Kernel 1: bucket batch rows by layer_stack index -> permutation lists
// ---------------------------------------------------------------------------
__global__ void nnue_bucket(const int* __restrict__ lsi,
                            int* __restrict__ counts,
                            int* __restrict__ perm) {
    int b = blockIdx.x * blockDim.x + threadIdx.x;
    if (b < B_SZ) {
        int s = lsi[b];
        int slot = atomicAdd(&counts[s], 1);
        perm[s * B_SZ + slot] = b;
    }
}

// ---------------------------------------------------------------------------
// Kernel 2: feature transform + perspective mix + LSQ quant + pair-product.
// One workgroup (256 threads) per batch element. Thread t owns columns
// [4t..4t+3] and [1024+4t..1024+4t+3] of both white and black accumulators,
// which is exactly what is needed to form l0_mixed[j] and l0_mixed[1024+j]
// with zero cross-thread traffic. All row loads are coalesced b128.
// ---------------------------------------------------------------------------
__device__ __forceinline__ float lsq_quant(float x, float rs, float s) {
    float g = x * rs;
    g = fminf(fmaxf(g, 0.0f), 255.0f);
    return rintf(g) * s;   // RNE, matches jnp.round
}

__global__ __launch_bounds__(256) void nnue_ft(
    const float* __restrict__ us,   const float* __restrict__ them,
    const int*   __restrict__ wIdx, const float* __restrict__ wVal,
    const int*   __restrict__ bIdx, const float* __restrict__ bVal,
    const int*   __restrict__ psqt, const float* __restrict__ ftW,
    const float* __restrict__ ftB,  const float* __restrict__ lsqS,
    float* __restrict__ l0m, float* __restrict__ partial)
{
    const int b = blockIdx.x;
    const int t = threadIdx.x;

    const float4* bias4 = (const float4*)ftB;
    float4 w0 = bias4[t], w1 = bias4[256 + t];   // white accum (cols j, 1024+j)
    float4 c0 = w0,       c1 = w1;               // black accum starts at same bias

    for (int k = 0; k < K_FT; ++k) {
        int   wi = wIdx[b * K_FT + k];
        float wv = wVal[b * K_FT + k];
        const float4* r4 = (const float4*)(ftW + (size_t)wi * D_FT);
        float4 a = r4[t], c = r4[256 + t];
        w0.x += wv * a.x; w0.y += wv * a.y; w0.z += wv * a.z; w0.w += wv * a.w;
        w1.x += wv * c.x; w1.y += wv * c.y; w1.z += wv * c.z; w1.w += wv * c.w;

        int   bi = bIdx[b * K_FT + k];
        float bv = bVal[b * K_FT + k];
        const float4* s4 = (const float4*)(ftW + (size_t)bi * D_FT);
        float4 e = s4[t], f = s4[256 + t];
        c0.x += bv * e.x; c0.y += bv * e.y; c0.z += bv * e.z; c0.w += bv * e.w;
        c1.x += bv * f.x; c1.y += bv * f.y; c1.z += bv * f.z; c1.w += bv * f.w;
    }

    const float u  = us[b];
    const float th = them[b];
    const float s0 = lsqS[0], s1 = lsqS[1], s2 = lsqS[2], s3 = lsqS[3];
    const float r0 = 1.0f / s0, r1 = 1.0f / s1, r2 = 1.0f / s2, r3 = 1.0f / s3;

    float4 m0, m1;
#define NNUE_COMP(c)                                                          \
    {                                                                         \
        float a0 = u * w0.c + th * c0.c;  /* s0: l0[j]        */              \
        float a1 = u * w1.c + th * c1.c;  /* s1: l0[1024+j]   */              \
        float a2 = u * c0.c + th * w0.c;  /* s2: l0[2048+j]   */              \
        float a3 = u * c1.c + th * w1.c;  /* s3: l0[3072+j]   */              \
        m0.c = lsq_quant(a0, r0, s0) * lsq_quant(a1, r1, s1);                 \
        m1.c = lsq_quant(a2, r2, s2) * lsq_quant(a3, r3, s3);                 \
    }
    NNUE_COMP(x) NNUE_COMP(y) NNUE_COMP(z) NNUE_COMP(w)
#undef NNUE_COMP

    float4* out4 = (float4*)(l0m + (size_t)b * L1);
    out4[t]       = m0;
    out4[256 + t] = m1;

    if (t == 0) {
        const int p = psqt[b];
        float wps = 0.0f, bps = 0.0f;   // bias cancels in (wps - bps)
        for (int k = 0; k < K_FT; ++k) {
            int   wi = wIdx[b * K_FT + k];
            float wv = wVal[b * K_FT + k];
            wps += wv * ftW[(size_t)wi * D_FT + 2048 + p];
            int   bi = bIdx[b * K_FT + k];
            float bv = bVal[b * K_FT + k];
            bps += bv * ftW[(size_t)bi * D_FT + 2048 + p];
        }
        partial[b] = (wps - bps) * (u - 0.5f);
    }
}

// ---------------------------------------------------------------------------
// Kernel 3: per-bucket MLP. One wave per 16-row tile. Layer 1 (2048->16)
// as V_WMMA_F32_16X16X4_F32 chain (M=16 batches, N=16 outputs, K=2048).
// Layers 2/3 via LDS + VALU inside the same wave.
//
// A layout (f32 16x4): lane holds M = lane%16, K = k0 + 2*(lane/16) + {0,1}.
// B layout (f32 4x16): lane holds N = lane%16, K = k0 + 2*(lane/16) + {0,1}.
// D layout: VGPR j = row j (lanes 0-15) / row j+8 (lanes 16-31), col = lane%16.
// ---------------------------------------------------------------------------
__global__ __launch_bounds__(32) void nnue_mlp(
    const float* __restrict__ l0m, const int* __restrict__ counts,
    const int* __restrict__ perm,  const float* __restrict__ partial,
    const float* __restrict__ W1,  const float* __restrict__ b1,
    const float* __restrict__ W2,  const float* __restrict__ b2,
    const float* __restrict__ Wo,  const float* __restrict__ bo,
    float* __restrict__ out)
{
    __shared__ float h1s[16][17];
    __shared__ float h2s[16][33];

    const int s    = blockIdx.y;
    const int tile = blockIdx.x;
    const int cnt  = counts[s];
    const int ntiles = (cnt + 15) >> 4;
    if (tile >= ntiles) return;               // uniform exit, EXEC stays all-1

    const int l    = threadIdx.x;
    const int n    = l & 15;                  // A row in tile == B column
    const int koff = (l >> 4) * 2;

    int rIdx   = tile * 16 + n;
    int rClamp = rIdx < cnt ? rIdx : cnt - 1; // pad partial tiles with a valid row
    int grow   = perm[s * B_SZ + rClamp];

    const float* aRow = l0m + (size_t)grow * L1;
    const float* bRow = W1 + ((size_t)s * 16 + n) * L1;   // W1[s][n][k]

    v8f acc = {};
    for (int k0 = 0; k0 < L1; k0 += 4) {
        v2f a  = *(const v2f*)(aRow + k0 + koff);
        v2f bb = *(const v2f*)(bRow + k0 + koff);
        acc = __builtin_amdgcn_wmma_f32_16x16x4_f32(
                  false, a, false, bb, (short)0, acc, false, false);
    }

    // Layer 1 epilogue: bias + clip, scatter D layout into LDS as h1[m][n]
    const float bias1  = b1[s * 16 + n];
    const int rowBase  = (l < 16) ? 0 : 8;
#pragma unroll
    for (int j = 0; j < 8; ++j) {
        float h = acc[j] + bias1;
        h1s[rowBase + j][n] = fminf(fmaxf(h, 0.0f), 1.0f);
    }
    __syncthreads();

    // Layer 2: 16 -> 32, lane l owns output o = l
    const float* w2row = W2 + ((size_t)s * 32 + l) * 16;
    const float  b2v   = b2[s * 32 + l];
    float w2r[16];
#pragma unroll
    for (int nn = 0; nn < 16; ++nn) w2r[nn] = w2row[nn];
    for (int m = 0; m < 16; ++m) {
        float a2 = b2v;
#pragma unroll
        for (int nn = 0; nn < 16; ++nn) a2 += w2r[nn] * h1s[m][nn];
        h2s[m][l] = fminf(fmaxf(a2, 0.0f), 1.0f);
    }
    __syncthreads();

    // Layer 3: 32 -> 1, lanes 0..15 own one batch row each
    if (l < 16) {
        int r = tile * 16 + l;
        if (r < cnt) {
            const float* worow = Wo + (size_t)s * 32;
            float a3 = bo[s];
#pragma unroll
            for (int o = 0; o < 32; ++o) a3 += worow[o] * h2s[l][o];
            int g = perm[s * B_SZ + r];
            out[g] = a3 + partial[g];
        }
    }
}

// ---------------------------------------------------------------------------
extern "C" void kernel_launch(void* const* d_in, const int* in_sizes, int n_in,
                              void* d_out, int out_size, void* d_ws, size_t ws_size,
                              hipStream_t stream) {
    (void)in_sizes; (void)n_in; (void)out_size; (void)ws_size;

    const float* us   = (const float*)d_in[0];
    const float* them = (const float*)d_in[1];
    const int*   wIdx = (const int*)  d_in[2];
    const float* wVal = (const float*)d_in[3];
    const int*   bIdx = (const int*)  d_in[4];
    const float* bVal = (const float*)d_in[5];
    const int*   psqt = (const int*)  d_in[6];
    const int*   lsi  = (const int*)  d_in[7];
    const float* ftW  = (const float*)d_in[8];
    const float* ftB  = (const float*)d_in[9];
    const float* lsqS = (const float*)d_in[10];
    const float* W1   = (const float*)d_in[11];
    const float* b1   = (const float*)d_in[12];
    const float* W2   = (const float*)d_in[13];
    const float* b2   = (const float*)d_in[14];
    const float* Wo   = (const float*)d_in[15];
    const float* bo   = (const float*)d_in[16];
    float* out = (float*)d_out;

    // scratch layout (all 256B aligned)
    char*  ws      = (char*)d_ws;
    int*   counts  = (int*)ws;                          //   256 B
    int*   perm    = (int*)(ws + 256);                  // 64 KB
    float* partial = (float*)(ws + 256 + 65536);        //  8 KB
    float* l0m     = (float*)(ws + 256 + 65536 + 8192); // 16 MB

    nnue_init_counts<<<1, 32, 0, stream>>>(counts);
    nnue_bucket<<<(B_SZ + 255) / 256, 256, 0, stream>>>(lsi, counts, perm);
    nnue_ft<<<B_SZ, 256, 0, stream>>>(us, them, wIdx, wVal, bIdx, bVal,
                                      psqt, ftW, ftB, lsqS, l0m, partial);
    dim3 grid(B_SZ / 16, NLS);
    nnue_mlp<<<grid, 32, 0, stream>>>(l0m, counts, perm, partial,
                                      W1, b1, W2, b2, Wo, bo, out);
}